// APPNP1BN_55121610277359
// MI455X (gfx1250) — compile-verified
//
#include <hip/hip_runtime.h>
#include <math.h>

typedef __attribute__((ext_vector_type(2))) float v2f;
typedef __attribute__((ext_vector_type(8))) float v8f;

#define N_NODES 100000
#define F_IN    256
#define HDIM    256
#define CDIM    64
#define BN_EPS  1e-5f
#define ALPHA   0.1f

// ---------------- utility ----------------
__global__ void k_zero(float* __restrict__ p, long n) {
  long i = (long)blockIdx.x * blockDim.x + threadIdx.x;
  long stride = (long)gridDim.x * blockDim.x;
  for (; i < n; i += stride) p[i] = 0.0f;
}

// deg[dst] += 1 per edge (float atomics; deg buffer pre-zeroed)
__global__ void k_count(const long long* __restrict__ dst, float* __restrict__ deg, int E) {
  int e = blockIdx.x * blockDim.x + threadIdx.x;
  if (e < E) atomicAdd(&deg[dst[e]], 1.0f);
}

// dinv = rsqrt(deg + 1)  (in place)
__global__ void k_dinv(float* __restrict__ d, int n) {
  int i = blockIdx.x * blockDim.x + threadIdx.x;
  if (i < n) d[i] = rsqrtf(d[i] + 1.0f);
}

// ---------------- WMMA GEMM: D[M x NCOLS] = A[M x K] * B[K x NCOLS] (+bias) ---
// Compile-time K/NCOLS -> all inner-loop addressing folds into immediate
// 24-bit VMEM offsets (max 252*256*4 = 258KB). Each wave computes TWO 16x16
// M-tiles sharing one B fragment (2 WMMAs per B load pair).
template <int K, int NCOLS, bool HAS_BIAS>
__global__ void k_gemm(const float* __restrict__ A, const float* __restrict__ B,
                       float* __restrict__ D, const float* __restrict__ bias) {
  const int lane  = threadIdx.x & 31;
  const int wave  = threadIdx.x >> 5;
  const int ntile = (NCOLS > 64) ? (blockIdx.y * 4 + wave) : wave;
  const size_t m0 = (size_t)blockIdx.x * 32;             // 2 M-tiles per wave
  const int colx  = lane & 15;
  const int col   = ntile * 16 + colx;
  const int kh    = (lane >> 4) * 2;                     // upper half-lanes: K+2,K+3

  const float* Ap0 = A + (m0 + colx) * (size_t)K + kh;   // A rows: M = lane&15
  const float* Ap1 = Ap0 + (size_t)16 * K;
  const float* Bp  = B + (size_t)kh * NCOLS + col;       // B rows striped per VGPR

  v8f acc0 = {}, acc1 = {};
#pragma unroll 8
  for (int k0 = 0; k0 < K; k0 += 4) {
    v2f a0 = *(const v2f*)(Ap0 + k0);                    // contiguous K pair
    v2f a1 = *(const v2f*)(Ap1 + k0);
    v2f b;
    b.x = Bp[(size_t)k0 * NCOLS];
    b.y = Bp[(size_t)(k0 + 1) * NCOLS];
    acc0 = __builtin_amdgcn_wmma_f32_16x16x4_f32(false, a0, false, b,
                                                 (short)0, acc0, false, false);
    acc1 = __builtin_amdgcn_wmma_f32_16x16x4_f32(false, a1, false, b,
                                                 (short)0, acc1, false, false);
  }
  // C/D layout: VGPR i -> M = i + (lane>=16)*8, N = lane&15
  const size_t rbase = m0 + (lane >> 4) * 8;
  float bb = 0.0f;
  if (HAS_BIAS) bb = bias[col];
#pragma unroll
  for (int i = 0; i < 8; ++i) {
    D[(rbase + i) * (size_t)NCOLS + col]      = acc0[i] + bb;
    D[(rbase + 16 + i) * (size_t)NCOLS + col] = acc1[i] + bb;
  }
}

// ---------------- edge scatter: agg[dst] += dinv[src]*dinv[dst] * h[src] -------
// One wave per edge; 256 floats per row = 2 x float4 per lane (coalesced).
// h fits in the 192MB L2, so gather + atomic traffic stays on-chip.
__global__ void k_scatter(const long long* __restrict__ src, const long long* __restrict__ dst,
                          const float* __restrict__ dinv, const float* __restrict__ h,
                          float* __restrict__ agg, int E) {
  int wid = blockIdx.x * (blockDim.x >> 5) + (threadIdx.x >> 5);
  if (wid >= E) return;
  int lane = threadIdx.x & 31;
  long long s = src[wid];
  long long d = dst[wid];
  float w = dinv[s] * dinv[d];
  const float4* hp = (const float4*)(h + (size_t)s * HDIM);
  float* ap = agg + (size_t)d * HDIM;
#pragma unroll
  for (int j = 0; j < 2; ++j) {
    float4 v = hp[lane + j * 32];
    int o = (lane + j * 32) * 4;
    atomicAdd(ap + o + 0, v.x * w);
    atomicAdd(ap + o + 1, v.y * w);
    atomicAdd(ap + o + 2, v.z * w);
    atomicAdd(ap + o + 3, v.w * w);
  }
}

// h = agg + dinv^2 * h + b1    (GCN self-loop + bias)
__global__ void k_post_gcn(float* __restrict__ h, const float* __restrict__ agg,
                           const float* __restrict__ dinv, const float* __restrict__ b1,
                           long total) {
  long i = (long)blockIdx.x * blockDim.x + threadIdx.x;
  if (i >= total) return;
  long row = i >> 8;
  int  col = (int)(i & 255);
  float di = dinv[row];
  h[i] = agg[i] + di * di * h[i] + b1[col];
}

// per-column sum / sumsq partials (256 threads = 256 columns, ROWS rows/block)
#define STAT_ROWS 256
__global__ void k_colstats(const float* __restrict__ h, float* __restrict__ sums,
                           float* __restrict__ sumsq, int n) {
  int col = threadIdx.x;
  int r0 = blockIdx.x * STAT_ROWS;
  int r1 = r0 + STAT_ROWS; if (r1 > n) r1 = n;
  float s = 0.0f, s2 = 0.0f;
  for (int r = r0; r < r1; ++r) {
    float v = h[(size_t)r * HDIM + col];
    s += v; s2 += v * v;
  }
  atomicAdd(&sums[col], s);
  atomicAdd(&sumsq[col], s2);
}

// batch-norm (biased var) + ReLU; writes both h and h0
__global__ void k_bn_relu(float* __restrict__ h, float* __restrict__ h0,
                          const float* __restrict__ sums, const float* __restrict__ sumsq,
                          const float* __restrict__ gamma, const float* __restrict__ beta,
                          long total) {
  long i = (long)blockIdx.x * blockDim.x + threadIdx.x;
  if (i >= total) return;
  int col = (int)(i & 255);
  const float inv_n = 1.0f / (float)N_NODES;
  float mean = sums[col] * inv_n;
  float var  = sumsq[col] * inv_n - mean * mean;
  float v = (h[i] - mean) * rsqrtf(var + BN_EPS) * gamma[col] + beta[col];
  v = fmaxf(v, 0.0f);
  h[i] = v;
  h0[i] = v;
}

// h = (1-alpha)*(agg + dinv^2*h) + alpha*h0
__global__ void k_appnp(float* __restrict__ h, const float* __restrict__ agg,
                        const float* __restrict__ dinv, const float* __restrict__ h0,
                        long total) {
  long i = (long)blockIdx.x * blockDim.x + threadIdx.x;
  if (i >= total) return;
  long row = i >> 8;
  float di = dinv[row];
  h[i] = (1.0f - ALPHA) * (agg[i] + di * di * h[i]) + ALPHA * h0[i];
}

// in-place row log-softmax over 64 logits; one wave per row, 2 values/lane (wave32)
__global__ void k_logsoftmax(float* __restrict__ out, int nrows) {
  int r = blockIdx.x * (blockDim.x >> 5) + (threadIdx.x >> 5);
  if (r >= nrows) return;
  int lane = threadIdx.x & 31;
  float* p = out + (size_t)r * CDIM;
  float a = p[lane];
  float b = p[lane + 32];
  float m = fmaxf(a, b);
#pragma unroll
  for (int off = 16; off > 0; off >>= 1) m = fmaxf(m, __shfl_xor(m, off, 32));
  float s = __expf(a - m) + __expf(b - m);
#pragma unroll
  for (int off = 16; off > 0; off >>= 1) s += __shfl_xor(s, off, 32);
  float ls = m + __logf(s);
  p[lane]      = a - ls;
  p[lane + 32] = b - ls;
}

// ---------------- host launch ----------------
extern "C" void kernel_launch(void* const* d_in, const int* in_sizes, int n_in,
                              void* d_out, int out_size, void* d_ws, size_t ws_size,
                              hipStream_t stream) {
  const float*     x     = (const float*)d_in[0];
  const long long* ei    = (const long long*)d_in[1];
  const float*     W1    = (const float*)d_in[2];
  const float*     b1    = (const float*)d_in[3];
  const float*     gamma = (const float*)d_in[4];
  const float*     beta  = (const float*)d_in[5];
  const float*     fcW   = (const float*)d_in[6];
  const float*     fcb   = (const float*)d_in[7];
  float*           out   = (float*)d_out;

  const int E = in_sizes[1] / 2;
  const long long* srcp = ei;
  const long long* dstp = ei + E;

  const long  NH  = (long)N_NODES * HDIM;      // 25.6M floats
  float* h     = (float*)d_ws;
  float* h0    = h   + NH;
  float* agg   = h0  + NH;
  float* dinv  = agg + NH;
  float* sums  = dinv + N_NODES;
  float* sumsq = sums + HDIM;

  const dim3 B256(256);
  const int  GElem  = (int)((NH + 255) / 256);             // 100000 blocks
  const int  GEdgeT = (E + 255) / 256;                     // thread-per-edge
  const int  GEdgeW = (E + 7) / 8;                         // wave-per-edge (8 waves/block)
  const int  GNode  = (N_NODES + 255) / 256;

  // degrees -> dinv
  hipLaunchKernelGGL(k_zero,  dim3(512),    B256, 0, stream, dinv, (long)N_NODES);
  hipLaunchKernelGGL(k_count, dim3(GEdgeT), B256, 0, stream, dstp, dinv, E);
  hipLaunchKernelGGL(k_dinv,  dim3(GNode),  B256, 0, stream, dinv, N_NODES);

  // h = x @ W1   (WMMA fp32; 3125 blocks of 32 rows x 4 y-blocks*4 waves = 16 N-tiles)
  hipLaunchKernelGGL((k_gemm<F_IN, HDIM, false>), dim3(N_NODES / 32, HDIM / 64), dim3(128),
                     0, stream, x, W1, h, (const float*)nullptr);

  // GCN propagate + bias
  hipLaunchKernelGGL(k_zero,     dim3(2048),  B256, 0, stream, agg, NH);
  hipLaunchKernelGGL(k_scatter,  dim3(GEdgeW), B256, 0, stream, srcp, dstp, dinv, h, agg, E);
  hipLaunchKernelGGL(k_post_gcn, dim3(GElem), B256, 0, stream, h, agg, dinv, b1, NH);

  // BatchNorm (training stats) + ReLU -> h, h0
  hipLaunchKernelGGL(k_zero,     dim3(1), B256, 0, stream, sums, (long)(2 * HDIM));
  hipLaunchKernelGGL(k_colstats, dim3((N_NODES + STAT_ROWS - 1) / STAT_ROWS), B256, 0, stream,
                     h, sums, sumsq, N_NODES);
  hipLaunchKernelGGL(k_bn_relu,  dim3(GElem), B256, 0, stream, h, h0, sums, sumsq, gamma, beta, NH);

  // APPNP: 10 propagation steps
  for (int k = 0; k < 10; ++k) {
    hipLaunchKernelGGL(k_zero,    dim3(2048),  B256, 0, stream, agg, NH);
    hipLaunchKernelGGL(k_scatter, dim3(GEdgeW), B256, 0, stream, srcp, dstp, dinv, h, agg, E);
    hipLaunchKernelGGL(k_appnp,   dim3(GElem), B256, 0, stream, h, agg, dinv, h0, NH);
  }

  // logits = h @ fcW + fcb   (WMMA fp32; 4 N-tiles = 4 waves = 1 block in y)
  hipLaunchKernelGGL((k_gemm<HDIM, CDIM, true>), dim3(N_NODES / 32, 1), dim3(128),
                     0, stream, h, fcW, out, fcb);

  // in-place log-softmax over 64 classes per row
  hipLaunchKernelGGL(k_logsoftmax, dim3((N_NODES + 7) / 8), B256, 0, stream, out, N_NODES);
}